// DRP_LAYER_463856468253
// MI455X (gfx1250) — compile-verified
//
#include <hip/hip_runtime.h>

typedef __attribute__((ext_vector_type(2))) float v2f;
typedef __attribute__((ext_vector_type(8))) float v8f;

#define CFL 0.5f
// Grid shapes
#define EHD 513          // E is (B, 513, 513)
#define HXR 511          // Hx is (B, 511, 512)
#define HXC 512
#define HYR 512          // Hy is (B, 512, 511)
#define HYC 511
#define NB  16

// ---------------------------------------------------------------------------
// Prologue: wf[r*4+c] = beta*fb + delta*fd + fy   (12 coefficients)
// ---------------------------------------------------------------------------
__global__ __launch_bounds__(32)
void drp_coeff_kernel(const float* __restrict__ beta, const float* __restrict__ delta,
                      const float* __restrict__ fb, const float* __restrict__ fd,
                      const float* __restrict__ fy, float* __restrict__ wf) {
    int t = threadIdx.x;
    if (t < 12) wf[t] = beta[0] * fb[t] + delta[0] * fd[t] + fy[t];
}

// ---------------------------------------------------------------------------
// WMMA boundary GEMMs.  D = A(16x4 f32) x B(4x16 f32) + C, K-chunked.
// A layout: lane L holds A[M=L&15][K=2*(L>>4)+v] in vgpr v.
// B layout: lane L holds B[K=2*(L>>4)+v][N=L&15] in vgpr v.
// C/D:      lane L holds D[M=v+8*(L>>4)][N=L&15] in vgpr v.
// ---------------------------------------------------------------------------

// Amper column terms from Hy:  C1[b, i<509, j<2, n<4] =
//   sum_{a<4, d<510} Hy[b, i+a, j+d] * K_n[d*4 + a]      (n: kf,kb,fu,fdn)
__global__ __launch_bounds__(32)
void amper_cols_wmma(const float* __restrict__ Hy,
                     const float* __restrict__ kf, const float* __restrict__ kb,
                     const float* __restrict__ fu, const float* __restrict__ fdn,
                     float* __restrict__ C1) {
    const int lane = threadIdx.x;
    const int tile = blockIdx.x & 63;
    const int b    = blockIdx.x >> 6;
    const int i0   = tile * 8;
    const int m    = lane & 15;
    const int hi   = lane >> 4;
    const int j    = m & 1;
    const int irow = i0 + (m >> 1);

    int r0 = irow + 2 * hi; if (r0 > HYR - 1) r0 = HYR - 1;   // clamp (discarded rows)
    int r1 = r0 + 1;        if (r1 > HYR - 1) r1 = HYR - 1;
    const float* row0 = Hy + ((size_t)b * HYR + r0) * HYC + j;
    const float* row1 = Hy + ((size_t)b * HYR + r1) * HYC + j;

    const float* kn = (m == 1) ? kb : (m == 2) ? fu : (m == 3) ? fdn : kf;
    const float  msk = (m < 4) ? 1.f : 0.f;
    const int    koff = 2 * hi;

    v8f acc = {};
    for (int d = 0; d < 510; ++d) {
        v2f a;  a.x = row0[d];               a.y = row1[d];
        v2f bb; bb.x = kn[d * 4 + koff] * msk; bb.y = kn[d * 4 + koff + 1] * msk;
        acc = __builtin_amdgcn_wmma_f32_16x16x4_f32(false, a, false, bb,
                                                    (short)0, acc, false, false);
    }
    if (m < 4) {
        #pragma unroll
        for (int v = 0; v < 8; ++v) {
            int mm = v + 8 * hi;
            int i  = i0 + (mm >> 1);
            int jj = mm & 1;
            if (i < 509)
                C1[(((size_t)b * 509 + i) * 2 + jj) * 4 + m] = acc[v];
        }
    }
}

// Amper row terms from Hx:  C2[b, i<2, xx<509, n<4] =
//   sum_{d<510, c<4} Hx[b, i+d, xx+c] * K_n[d*4 + c]
__global__ __launch_bounds__(32)
void amper_rows_wmma(const float* __restrict__ Hx,
                     const float* __restrict__ kf, const float* __restrict__ kb,
                     const float* __restrict__ fu, const float* __restrict__ fdn,
                     float* __restrict__ C2) {
    const int lane = threadIdx.x;
    const int tile = blockIdx.x & 63;
    const int b    = blockIdx.x >> 6;
    const int xx0  = tile * 8;
    const int m    = lane & 15;
    const int hi   = lane >> 4;
    const int iy   = m & 1;
    const int xxm  = xx0 + (m >> 1);

    int c0 = xxm + 2 * hi; if (c0 > HXC - 2) c0 = HXC - 2;    // keep c0+1 in range
    const float* base = Hx + (size_t)b * HXR * HXC + (size_t)iy * HXC + c0;

    const float* kn = (m == 1) ? kb : (m == 2) ? fu : (m == 3) ? fdn : kf;
    const float  msk = (m < 4) ? 1.f : 0.f;
    const int    koff = 2 * hi;

    v8f acc = {};
    for (int d = 0; d < 510; ++d) {
        const float* p = base + (size_t)d * HXC;
        v2f a;  a.x = p[0];                  a.y = p[1];
        v2f bb; bb.x = kn[d * 4 + koff] * msk; bb.y = kn[d * 4 + koff + 1] * msk;
        acc = __builtin_amdgcn_wmma_f32_16x16x4_f32(false, a, false, bb,
                                                    (short)0, acc, false, false);
    }
    if (m < 4) {
        #pragma unroll
        for (int v = 0; v < 8; ++v) {
            int mm = v + 8 * hi;
            int xx = xx0 + (mm >> 1);
            int ii = mm & 1;
            if (xx < 509)
                C2[(((size_t)b * 2 + ii) * 509 + xx) * 4 + m] = acc[v];
        }
    }
}

// Faraday column terms from E:  C3[b, i<511, j<2, n<2] =
//   sum_{r<3, d<512} E[b, i+r, j+d] * K_n[r*512 + d]       (n: kef, keb)
__global__ __launch_bounds__(32)
void far_cols_wmma(const float* __restrict__ E,
                   const float* __restrict__ kef, const float* __restrict__ keb,
                   float* __restrict__ C3) {
    const int lane = threadIdx.x;
    const int tile = blockIdx.x & 63;
    const int b    = blockIdx.x >> 6;
    const int i0   = tile * 8;
    const int m    = lane & 15;
    const int hi   = lane >> 4;
    const int j    = m & 1;
    const int irow = i0 + (m >> 1);

    const float* kn = (m == 1) ? keb : kef;
    const float  msk = (m < 2) ? 1.f : 0.f;

    v8f acc = {};
    for (int r = 0; r < 3; ++r) {
        int rr = irow + r; if (rr > EHD - 1) rr = EHD - 1;
        const float* erow = E + ((size_t)b * EHD + rr) * EHD + j + 2 * hi;
        const float* krow = kn + r * 512 + 2 * hi;
        for (int d0 = 0; d0 < 512; d0 += 4) {
            v2f a;  a.x = erow[d0];         a.y = erow[d0 + 1];
            v2f bb; bb.x = krow[d0] * msk;  bb.y = krow[d0 + 1] * msk;
            acc = __builtin_amdgcn_wmma_f32_16x16x4_f32(false, a, false, bb,
                                                        (short)0, acc, false, false);
        }
    }
    if (m < 2) {
        #pragma unroll
        for (int v = 0; v < 8; ++v) {
            int mm = v + 8 * hi;
            int i  = i0 + (mm >> 1);
            int jj = mm & 1;
            if (i < 511)
                C3[(((size_t)b * 511 + i) * 2 + jj) * 2 + m] = acc[v];
        }
    }
}

// Faraday row terms from E:  C4[b, y<2, jj<511, n<2] =
//   sum_{c<3, d<512} E[b, y+d, jj+c] * K_n[c*512 + d]
__global__ __launch_bounds__(32)
void far_rows_wmma(const float* __restrict__ E,
                   const float* __restrict__ kef, const float* __restrict__ keb,
                   float* __restrict__ C4) {
    const int lane = threadIdx.x;
    const int tile = blockIdx.x & 63;
    const int b    = blockIdx.x >> 6;
    const int j0   = tile * 8;
    const int m    = lane & 15;
    const int hi   = lane >> 4;
    const int y    = m & 1;
    const int jjm  = j0 + (m >> 1);

    const float* kn = (m == 1) ? keb : kef;
    const float  msk = (m < 2) ? 1.f : 0.f;

    v8f acc = {};
    for (int c = 0; c < 3; ++c) {
        int col = jjm + c; if (col > EHD - 1) col = EHD - 1;
        const float* ecol = E + (size_t)b * EHD * EHD + col;
        const float* krow = kn + c * 512 + 2 * hi;
        for (int d0 = 0; d0 < 512; d0 += 4) {
            int r0 = y + d0 + 2 * hi;                      // <= 512, in range
            v2f a;  a.x = ecol[(size_t)r0 * EHD]; a.y = ecol[(size_t)(r0 + 1) * EHD];
            v2f bb; bb.x = krow[d0] * msk;        bb.y = krow[d0 + 1] * msk;
            acc = __builtin_amdgcn_wmma_f32_16x16x4_f32(false, a, false, bb,
                                                        (short)0, acc, false, false);
        }
    }
    if (m < 2) {
        #pragma unroll
        for (int v = 0; v < 8; ++v) {
            int mm = v + 8 * hi;
            int jj = j0 + (mm >> 1);
            int yy = mm & 1;
            if (jj < 511)
                C4[(((size_t)b * 2 + yy) * 511 + jj) * 2 + m] = acc[v];
        }
    }
}

// ---------------------------------------------------------------------------
// Bulk kernels (12-tap interior stencils, L2-resident) + boundary injection.
// ---------------------------------------------------------------------------
__global__ __launch_bounds__(256)
void amper_main(const float* __restrict__ E, const float* __restrict__ Hx,
                const float* __restrict__ Hy, const float* __restrict__ wf,
                const float* __restrict__ C1, const float* __restrict__ C2,
                float* __restrict__ Eout) {
    int idx = blockIdx.x * 256 + threadIdx.x;
    if (idx >= NB * EHD * EHD) return;
    int b   = idx / (EHD * EHD);
    int rem = idx - b * (EHD * EHD);
    int y   = rem / EHD;
    int x   = rem - y * EHD;

    float w[12];
    #pragma unroll
    for (int t = 0; t < 12; ++t) w[t] = wf[t];

    float acc = 0.f;
    const float* hyb = Hy + (size_t)b * HYR * HYC;
    const float* hxb = Hx + (size_t)b * HXR * HXC;
    bool yin = (y >= 2) && (y <= 510);
    bool xin = (x >= 2) && (x <= 510);

    if (yin && xin) {
        float s1 = 0.f, s2 = 0.f;
        #pragma unroll
        for (int c = 0; c < 3; ++c)
            #pragma unroll
            for (int a = 0; a < 4; ++a)
                s1 += hyb[(size_t)(y - 2 + a) * HYC + (x - 2 + c)] * w[c * 4 + a];
        #pragma unroll
        for (int r = 0; r < 3; ++r)
            #pragma unroll
            for (int c = 0; c < 4; ++c)
                s2 += hxb[(size_t)(y - 2 + r) * HXC + (x - 2 + c)] * w[r * 4 + c];
        acc = s1 - s2;
    }
    if (yin) {   // s1 boundary columns (n: 0=kf,1=kb,2=fu,3=fdn)
        const float* c1 = C1 + ((size_t)b * 509 + (y - 2)) * 8;   // [j*4+n]
        if (x == 0)        acc += c1[0];
        else if (x == 1)   acc += c1[4] + c1[2];
        else if (x == 2)   acc += c1[6];
        if (x == 510)      acc += c1[3];
        else if (x == 511) acc += c1[1] + c1[7];
        else if (x == 512) acc += c1[5];
    }
    if (xin) {   // s2 boundary rows (subtract)
        const float* c2  = C2 + (size_t)b * 2 * 509 * 4 + (size_t)(x - 2) * 4;
        const float* c2b = c2 + 509 * 4;
        if (y == 0)        acc -= c2[0];
        else if (y == 1)   acc -= c2b[0] + c2[2];
        else if (y == 2)   acc -= c2b[2];
        if (y == 510)      acc -= c2[3];
        else if (y == 511) acc -= c2[1] + c2b[3];
        else if (y == 512) acc -= c2b[1];
    }
    Eout[idx] = E[idx] + CFL * acc;
}

__global__ __launch_bounds__(256)
void far_hx_main(const float* __restrict__ E, const float* __restrict__ Hx,
                 const float* __restrict__ wf, const float* __restrict__ C3,
                 float* __restrict__ Hxout) {
    int idx = blockIdx.x * 256 + threadIdx.x;
    if (idx >= NB * HXR * HXC) return;
    int b   = idx / (HXR * HXC);
    int rem = idx - b * (HXR * HXC);
    int i   = rem / HXC;
    int x   = rem - i * HXC;

    float w[12];
    #pragma unroll
    for (int t = 0; t < 12; ++t) w[t] = wf[t];

    const float* eb = E + (size_t)b * EHD * EHD;
    float s = 0.f;
    if (x >= 1 && x <= 510) {
        #pragma unroll
        for (int r = 0; r < 3; ++r)
            #pragma unroll
            for (int c = 0; c < 4; ++c)
                s += eb[(size_t)(i + r) * EHD + (x - 1 + c)] * w[r * 4 + c];
    }
    const float* c3 = C3 + ((size_t)b * 511 + i) * 4;   // [j*2+n], n: 0=kef,1=keb
    if (x == 0)        s += c3[0];
    else if (x == 1)   s += c3[2];
    if (x == 510)      s += c3[1];
    else if (x == 511) s += c3[3];
    Hxout[idx] = Hx[idx] - CFL * s;
}

__global__ __launch_bounds__(256)
void far_hy_main(const float* __restrict__ E, const float* __restrict__ Hy,
                 const float* __restrict__ wf, const float* __restrict__ C4,
                 float* __restrict__ Hyout) {
    int idx = blockIdx.x * 256 + threadIdx.x;
    if (idx >= NB * HYR * HYC) return;
    int b   = idx / (HYR * HYC);
    int rem = idx - b * (HYR * HYC);
    int y   = rem / HYC;
    int j   = rem - y * HYC;

    float w[12];
    #pragma unroll
    for (int t = 0; t < 12; ++t) w[t] = wf[t];

    const float* eb = E + (size_t)b * EHD * EHD;
    float s = 0.f;
    if (y >= 1 && y <= 510) {
        #pragma unroll
        for (int a = 0; a < 4; ++a)
            #pragma unroll
            for (int c = 0; c < 3; ++c)
                s += eb[(size_t)(y - 1 + a) * EHD + (j + c)] * w[c * 4 + a];
    }
    const float* c4 = C4 + (size_t)b * 2 * 511 * 2;     // [yy*511*2 + jj*2 + n]
    if (y == 0)        s += c4[(size_t)j * 2 + 0];
    else if (y == 1)   s += c4[(size_t)(511 + j) * 2 + 0];
    if (y == 510)      s += c4[(size_t)j * 2 + 1];
    else if (y == 511) s += c4[(size_t)(511 + j) * 2 + 1];
    Hyout[idx] = Hy[idx] + CFL * s;
}

// ---------------------------------------------------------------------------
extern "C" void kernel_launch(void* const* d_in, const int* in_sizes, int n_in,
                              void* d_out, int out_size, void* d_ws, size_t ws_size,
                              hipStream_t stream) {
    (void)in_sizes; (void)n_in; (void)out_size; (void)ws_size;

    const float* E1    = (const float*)d_in[0];
    const float* Hx1   = (const float*)d_in[1];
    const float* Hy1   = (const float*)d_in[2];
    const float* beta  = (const float*)d_in[9];
    const float* delta = (const float*)d_in[10];
    const float* fb    = (const float*)d_in[11];
    const float* fd    = (const float*)d_in[12];
    const float* fy    = (const float*)d_in[13];
    const float* kf    = (const float*)d_in[14];
    const float* kb    = (const float*)d_in[15];
    const float* fu    = (const float*)d_in[16];
    const float* fdn   = (const float*)d_in[17];
    const float* kef   = (const float*)d_in[18];
    const float* keb   = (const float*)d_in[19];

    float* ws = (float*)d_ws;
    float* wf = ws;
    float* C1 = ws + 16;                        // 16*509*2*4 = 65152
    float* C2 = C1 + 16 * 509 * 2 * 4;          // 65152
    float* C3 = C2 + 16 * 509 * 2 * 4;          // 16*511*2*2 = 32704
    float* C4 = C3 + 16 * 511 * 2 * 2;          // 32704

    float* out = (float*)d_out;
    const size_t NE  = (size_t)NB * EHD * EHD;
    const size_t NHX = (size_t)NB * HXR * HXC;
    const size_t NHY = (size_t)NB * HYR * HYC;
    float* E2 = out;        float* Hx2 = E2 + NE;  float* Hy2 = Hx2 + NHX;
    float* E3 = Hy2 + NHY;  float* Hx3 = E3 + NE;  float* Hy3 = Hx3 + NHX;
    float* E4 = Hy3 + NHY;  float* Hx4 = E4 + NE;  float* Hy4 = Hx4 + NHX;

    drp_coeff_kernel<<<1, 32, 0, stream>>>(beta, delta, fb, fd, fy, wf);

    const int GE  = (int)((NE  + 255) / 256);
    const int GHX = (int)((NHX + 255) / 256);
    const int GHY = (int)((NHY + 255) / 256);

    struct Rnd { const float* Ei; const float* Hxi; const float* Hyi;
                 float* Eo; float* Hxo; float* Hyo; };
    Rnd rounds[3] = { { E1, Hx1, Hy1, E2, Hx2, Hy2 },
                      { E2, Hx2, Hy2, E3, Hx3, Hy3 },
                      { E3, Hx3, Hy3, E4, Hx4, Hy4 } };

    for (int r = 0; r < 3; ++r) {
        Rnd q = rounds[r];
        amper_cols_wmma<<<NB * 64, 32, 0, stream>>>(q.Hyi, kf, kb, fu, fdn, C1);
        amper_rows_wmma<<<NB * 64, 32, 0, stream>>>(q.Hxi, kf, kb, fu, fdn, C2);
        amper_main<<<GE, 256, 0, stream>>>(q.Ei, q.Hxi, q.Hyi, wf, C1, C2, q.Eo);
        far_cols_wmma<<<NB * 64, 32, 0, stream>>>(q.Eo, kef, keb, C3);
        far_rows_wmma<<<NB * 64, 32, 0, stream>>>(q.Eo, kef, keb, C4);
        far_hx_main<<<GHX, 256, 0, stream>>>(q.Eo, q.Hxi, wf, C3, q.Hxo);
        far_hy_main<<<GHY, 256, 0, stream>>>(q.Eo, q.Hyi, wf, C4, q.Hyo);
    }
}